// BinLoss_11132555231951
// MI455X (gfx1250) — compile-verified
//
#include <hip/hip_runtime.h>
#include <hip/hip_bf16.h>

// Problem geometry (fixed by the reference).
#define NCH        88          // B*C = 8*11
#define NVOX       262144      // 64^3
#define BINS       20
#define NTOT       (NCH * (size_t)NVOX)     // 23,068,672 elements per tensor
#define BLOCKS_PER_CH 32
#define ELEMS_PER_BLOCK (NVOX / BLOCKS_PER_CH)   // 8192
#define THREADS    256
#define PASSES     (ELEMS_PER_BLOCK / (THREADS * 4))  // 8 float4 passes
#define NBLOCKS    (NCH * BLOCKS_PER_CH)    // 2816

typedef float v2f __attribute__((ext_vector_type(2)));
typedef float v8f __attribute__((ext_vector_type(8)));

__device__ __forceinline__ float smooth_l1_term(float a, float b) {
    float d = fabsf(a - b);
    return (d < 1.0f) ? (0.5f * d * d) : (d - 0.5f);
}

// Phase A of branchless binning: arithmetic candidate bin.
// g = floor((x+4)*2.5) is provably within +/-1 of the true bin (edges are a
// float32 linspace; |t_err| <= ~3e-6 while bins are 0.4 wide).
__device__ __forceinline__ int bin_candidate(float x) {
    int g = (int)floorf((x + 4.0f) * 2.5f);
    return min(max(g, 0), BINS - 1);
}

// Phase B: one +/-1 correction against the REAL (lo,hi) edges (exact,
// including values landing exactly on an edge), then one unconditional
// ds_add_u32. Out-of-range -> trash slot BINS.
__device__ __forceinline__ void bin_resolve_add(float x, int c, float2 e,
                                                unsigned int* __restrict__ h) {
    int idx = c + (int)(x >= e.y) - (int)(x < e.x);
    unsigned int u = (unsigned int)idx;
    atomicAdd(&h[u <= (BINS - 1) ? idx : BINS], 1u);
}

__global__ void __launch_bounds__(THREADS)
bin_loss_main(const float* __restrict__ inp, const float* __restrict__ tar,
              const float* __restrict__ bin_range,
              unsigned int* __restrict__ histA, unsigned int* __restrict__ histB,
              float* __restrict__ partial) {
    __shared__ float2 ep[BINS];                // (lo,hi) pairs, = bin_range rows
    __shared__ unsigned int hA[BINS + 1];      // +1 trash slot
    __shared__ unsigned int hB[BINS + 1];
    __shared__ float waveSum[THREADS / 32];

    const int tid = threadIdx.x;
    if (tid < BINS) {
        ep[tid] = ((const float2*)bin_range)[tid];
        hA[tid] = 0u;
        hB[tid] = 0u;
    }
    if (tid == BINS) { hA[BINS] = 0u; hB[BINS] = 0u; }
    __syncthreads();

    const int ch  = blockIdx.x / BLOCKS_PER_CH;
    const int sub = blockIdx.x % BLOCKS_PER_CH;
    const size_t base = (size_t)ch * NVOX + (size_t)sub * ELEMS_PER_BLOCK;

    const float4* __restrict__ ip = (const float4*)(inp + base);
    const float4* __restrict__ tp = (const float4*)(tar + base);

    // WMMA-based accumulation: C(16x16 f32) += A(16x4, smooth-L1 terms) x ones(4x16).
    // Each issue folds 128 per-wave terms (as 64 pair sums) into the accumulator via the matrix pipe.
    v8f acc = {};
    v2f onesB;
    onesB.x = 1.0f;
    onesB.y = 1.0f;

#pragma unroll
    for (int p = 0; p < PASSES; ++p) {
        if (p + 1 < PASSES) {
            __builtin_prefetch(&ip[(p + 1) * THREADS + tid], 0, 0);
            __builtin_prefetch(&tp[(p + 1) * THREADS + tid], 0, 0);
        }
        float4 a = ip[p * THREADS + tid];   // global_load_b128
        float4 b = tp[p * THREADS + tid];

        // ---- Phase A: batch ALL edge-pair LDS loads before any atomics ----
        // (keeps s_wait_dscnt partial instead of 0: edge-load consumers don't
        //  serialize on outstanding histogram ds_add atomics)
        float va[4] = {a.x, a.y, a.z, a.w};
        float vb[4] = {b.x, b.y, b.z, b.w};
        int    ca[4], cb[4];
        float2 ea[4], eb[4];
#pragma unroll
        for (int j = 0; j < 4; ++j) { ca[j] = bin_candidate(va[j]); ea[j] = ep[ca[j]]; }
#pragma unroll
        for (int j = 0; j < 4; ++j) { cb[j] = bin_candidate(vb[j]); eb[j] = ep[cb[j]]; }

        // ---- SmoothL1 + WMMA while the LDS loads are in flight ----
        float t0 = smooth_l1_term(a.x, b.x);
        float t1 = smooth_l1_term(a.y, b.y);
        float t2 = smooth_l1_term(a.z, b.z);
        float t3 = smooth_l1_term(a.w, b.w);

        v2f afrag;
        afrag.x = t0 + t1;
        afrag.y = t2 + t3;
        // D = A x ones + C  (all lanes active; uniform control flow here)
        acc = __builtin_amdgcn_wmma_f32_16x16x4_f32(
            /*neg_a=*/false, afrag, /*neg_b=*/false, onesB,
            /*c_mod=*/(short)0, acc, /*reuse_a=*/false, /*reuse_b=*/false);

        // ---- Phase B: resolve +/-1 and fire unconditional LDS atomics ----
#pragma unroll
        for (int j = 0; j < 4; ++j) bin_resolve_add(va[j], ca[j], ea[j], hA);
#pragma unroll
        for (int j = 0; j < 4; ++j) bin_resolve_add(vb[j], cb[j], eb[j], hB);
    }

    // Reduce the 16x16 accumulator. With B = all-ones every column holds the
    // same row-sums; column 0 lives in lane 0 (rows 0-7) and lane 16 (rows 8-15):
    // waveTotal = shfl(sum8, 0) + shfl(sum8, 16).
    float s8 = acc[0] + acc[1] + acc[2] + acc[3] + acc[4] + acc[5] + acc[6] + acc[7];
    float waveTot = __shfl(s8, 0, 32) + __shfl(s8, 16, 32);

    const int lane = tid & 31;
    const int wave = tid >> 5;
    if (lane == 0) waveSum[wave] = waveTot;
    __syncthreads();

    if (tid == 0) {
        float bs = 0.0f;
#pragma unroll
        for (int w = 0; w < THREADS / 32; ++w) bs += waveSum[w];
        partial[blockIdx.x] = bs;   // deterministic per-block partial, no float atomics
    }

    // Flush block-private histograms (20 global integer atomics per tensor).
    if (tid < BINS) {
        if (hA[tid]) atomicAdd(&histA[ch * BINS + tid], hA[tid]);
        if (hB[tid]) atomicAdd(&histB[ch * BINS + tid], hB[tid]);
    }
}

__global__ void __launch_bounds__(THREADS)
bin_loss_final(const unsigned int* __restrict__ histA,
               const unsigned int* __restrict__ histB,
               const float* __restrict__ partial,
               float* __restrict__ out) {
    __shared__ float s1[THREADS];
    __shared__ float s2[THREADS];
    const int tid = threadIdx.x;

    float l1 = 0.0f;
    for (int i = tid; i < NBLOCKS; i += THREADS) l1 += partial[i];

    float hd = 0.0f;
    for (int i = tid; i < NCH * BINS; i += THREADS) {
        int d = (int)histA[i] - (int)histB[i];
        hd += fabsf((float)d);
    }

    s1[tid] = l1;
    s2[tid] = hd;
    __syncthreads();
    for (int off = THREADS / 2; off > 0; off >>= 1) {
        if (tid < off) {
            s1[tid] += s1[tid + off];
            s2[tid] += s2[tid + off];
        }
        __syncthreads();
    }
    if (tid == 0) {
        float loss1 = s1[0] / (float)NTOT;
        float loss2 = s2[0] / ((float)NVOX * (float)(NCH * BINS));
        out[0] = 0.5f * loss1 + 0.5f * loss2;
    }
}

extern "C" void kernel_launch(void* const* d_in, const int* in_sizes, int n_in,
                              void* d_out, int out_size, void* d_ws, size_t ws_size,
                              hipStream_t stream) {
    const float* inp = (const float*)d_in[0];   // [8,11,64,64,64] f32
    const float* tar = (const float*)d_in[1];   // [8,11,64,64,64] f32
    const float* br  = (const float*)d_in[2];   // [20,2] f32

    unsigned int* histA = (unsigned int*)d_ws;
    unsigned int* histB = histA + NCH * BINS;
    float* partial = (float*)(histB + NCH * BINS);

    // Zero only the histogram counters (capture-safe memset node).
    hipMemsetAsync(d_ws, 0, 2 * NCH * BINS * sizeof(unsigned int), stream);

    bin_loss_main<<<NBLOCKS, THREADS, 0, stream>>>(inp, tar, br, histA, histB, partial);
    bin_loss_final<<<1, THREADS, 0, stream>>>(histA, histB, partial, (float*)d_out);
}